// AttLayer_model_5_46755013984607
// MI455X (gfx1250) — compile-verified
//
#include <hip/hip_runtime.h>
#include <hip/hip_bf16.h>
#include <math.h>

#define BB 64
#define TT 2048
#define DD 256
#define HH 5
#define SCH 8            // T-chunks for the weighted-sum pass

typedef float v2f __attribute__((ext_vector_type(2)));
typedef float v8f __attribute__((ext_vector_type(8)));

// ---------------------------------------------------------------------------
// Kernel 1: scores[b,t] = sum_h tanh(x.Wt[:,h]+bt[h]) * tanh(xf*Wf[h]+bf[h]) * v[h] + sum(b)
// Projection x_temp(16xK) @ Wpad(Kx16) done with V_WMMA_F32_16X16X4_F32.
// One wave per 16-row tile; 4 waves per block; B*T/16 = 8192 tiles.
// ---------------------------------------------------------------------------
__global__ __launch_bounds__(128) void score_kernel(
    const float* __restrict__ X,    // (B,T,D)
    const float* __restrict__ xfea, // (B,T)
    const float* __restrict__ Wt,   // (D,H)
    const float* __restrict__ bt,   // (H)
    const float* __restrict__ Wf,   // (1,H)
    const float* __restrict__ bf,   // (H)
    const float* __restrict__ bvec, // (H)
    const float* __restrict__ uw,   // (H,H)
    float* __restrict__ scores)     // (B,T)
{
    const int lane = threadIdx.x & 31;
    const int wid  = threadIdx.x >> 5;
    const int tile = blockIdx.x * 4 + wid;      // 0 .. 8191
    const int b    = tile >> 7;                 // T/16 = 128 tiles per batch row
    const int t0   = (tile & 127) << 4;

    const int  n      = lane & 15;              // N column (0..15)
    const int  hi     = lane >> 4;              // lane half
    const int  nc     = (n < HH) ? n : (HH - 1);
    const bool active = (n < HH);

    // ---- WMMA accumulation: C(16x16) += A(16x4) * Bpad(4x16), K = 0..255 ----
    v8f c = {0.f,0.f,0.f,0.f,0.f,0.f,0.f,0.f};
    const size_t rowBase = ((size_t)b * TT + (size_t)(t0 + n)) * DD;

    #pragma unroll 4
    for (int k0 = 0; k0 < DD; k0 += 4) {
        const int k = k0 + 2 * hi;
        // A fragment: lane holds X[row, k] and X[row, k+1]  (8B aligned)
        const float2 av = *(const float2*)(X + rowBase + k);
        v2f a; a[0] = av.x; a[1] = av.y;
        // B fragment: lane holds Wpad[k, n], Wpad[k+1, n]; cols >= H are zero.
        float b0 = Wt[k * HH + nc];
        float b1 = Wt[(k + 1) * HH + nc];
        v2f bb; bb[0] = active ? b0 : 0.f; bb[1] = active ? b1 : 0.f;
        c = __builtin_amdgcn_wmma_f32_16x16x4_f32(
                false, a, false, bb, (short)0, c, false, false);
    }

    // ---- per-lane epilogue params (column = n) ----
    const float bt_n = bt[nc];
    const float wf_n = Wf[nc];
    const float bf_n = bf[nc];
    float v_n = 0.f;
    #pragma unroll
    for (int j = 0; j < HH; ++j) v_n += uw[nc * HH + j];  // row-sum of uw
    float cb = 0.f;
    #pragma unroll
    for (int j = 0; j < HH; ++j) cb += bvec[j];           // sum(b) (softmax-invariant)

    // C VGPR r holds rows m=r (lanes 0-15) and m=r+8 (lanes 16-31), col N=n.
    #pragma unroll
    for (int r = 0; r < 8; ++r) {
        const int trow = t0 + r + 8 * hi;
        const float xf = xfea[(size_t)b * TT + trow];
        float val = tanhf(c[r] + bt_n) * tanhf(xf * wf_n + bf_n) * v_n;
        val = active ? val : 0.f;
        // reduce across the 16-lane half (cols) -> row score at n==0
        val += __shfl_xor(val, 8, 16);
        val += __shfl_xor(val, 4, 16);
        val += __shfl_xor(val, 2, 16);
        val += __shfl_xor(val, 1, 16);
        if (n == 0) scores[(size_t)b * TT + trow] = val + cb;
    }
}

// ---------------------------------------------------------------------------
// Kernel 2: masked softmax over T (in place). One block per batch row.
// w_t = m_t * exp(s_t - max_all) / sum(m * exp(s - max_all))
// ---------------------------------------------------------------------------
__global__ __launch_bounds__(256) void softmax_kernel(
    const unsigned char* __restrict__ mask,  // (B,T) bool
    float* __restrict__ scores)              // (B,T) in: scores, out: weights
{
    __shared__ float red[256];
    const int b   = blockIdx.x;
    const int tid = threadIdx.x;

    float s[TT / 256];
    float m = -INFINITY;
    #pragma unroll
    for (int i = 0; i < TT / 256; ++i) {
        s[i] = scores[(size_t)b * TT + i * 256 + tid];
        m = fmaxf(m, s[i]);
    }
    red[tid] = m;
    __syncthreads();
    for (int off = 128; off > 0; off >>= 1) {
        if (tid < off) red[tid] = fmaxf(red[tid], red[tid + off]);
        __syncthreads();
    }
    const float M = red[0];
    __syncthreads();

    float e[TT / 256];
    float sum = 0.f;
    #pragma unroll
    for (int i = 0; i < TT / 256; ++i) {
        const float mk = mask[(size_t)b * TT + i * 256 + tid] ? 1.f : 0.f;
        e[i] = mk * expf(s[i] - M);
        sum += e[i];
    }
    red[tid] = sum;
    __syncthreads();
    for (int off = 128; off > 0; off >>= 1) {
        if (tid < off) red[tid] += red[tid + off];
        __syncthreads();
    }
    const float inv = 1.f / red[0];
    #pragma unroll
    for (int i = 0; i < TT / 256; ++i)
        scores[(size_t)b * TT + i * 256 + tid] = e[i] * inv;
}

// ---------------------------------------------------------------------------
// Kernel 3: partial weighted sums. Grid (SCH, B); thread d = tid.
// Weights for the chunk staged in LDS; x_temp read fully coalesced (1 KiB/row).
// ---------------------------------------------------------------------------
__global__ __launch_bounds__(256) void wsum_kernel(
    const float* __restrict__ X,      // (B,T,D)
    const float* __restrict__ w,      // (B,T)
    float* __restrict__ partial)      // (SCH, B, D)
{
    __shared__ float sw[TT / SCH];    // 256
    const int b   = blockIdx.y;
    const int s   = blockIdx.x;
    const int tid = threadIdx.x;
    const int t0  = s * (TT / SCH);

    sw[tid] = w[(size_t)b * TT + t0 + tid];
    __syncthreads();

    float acc = 0.f;
    const float* xp = X + ((size_t)b * TT + t0) * DD + tid;
    for (int t = 0; t < TT / SCH; ++t)
        acc += sw[t] * xp[(size_t)t * DD];

    partial[((size_t)s * BB + b) * DD + tid] = acc;
}

// ---------------------------------------------------------------------------
// Kernel 4: fold partials -> out (B,D)
// ---------------------------------------------------------------------------
__global__ __launch_bounds__(256) void reduce_kernel(
    const float* __restrict__ partial,  // (SCH, B, D)
    float* __restrict__ out)            // (B, D)
{
    const int id = blockIdx.x * 256 + threadIdx.x;   // 0 .. B*D-1
    float acc = 0.f;
    #pragma unroll
    for (int s = 0; s < SCH; ++s)
        acc += partial[(size_t)s * (BB * DD) + id];
    out[id] = acc;
}

extern "C" void kernel_launch(void* const* d_in, const int* in_sizes, int n_in,
                              void* d_out, int out_size, void* d_ws, size_t ws_size,
                              hipStream_t stream) {
    const float*         x_temp = (const float*)d_in[0];
    const float*         x_fea  = (const float*)d_in[1];
    const unsigned char* mask   = (const unsigned char*)d_in[2];
    const float*         W_temp = (const float*)d_in[3];
    const float*         b_temp = (const float*)d_in[4];
    const float*         W_fea  = (const float*)d_in[5];
    const float*         b_fea  = (const float*)d_in[6];
    const float*         bvec   = (const float*)d_in[7];
    const float*         uw     = (const float*)d_in[8];
    float*               out    = (float*)d_out;

    float* wsf     = (float*)d_ws;
    float* scores  = wsf;                       // B*T floats (512 KiB)
    float* partial = wsf + (size_t)BB * TT;     // SCH*B*D floats (512 KiB)

    // 1) scores via WMMA projection: 8192 wave-tiles, 4 waves/block
    score_kernel<<<(BB * TT / 16) / 4, 128, 0, stream>>>(
        x_temp, x_fea, W_temp, b_temp, W_fea, b_fea, bvec, uw, scores);

    // 2) masked softmax per batch row
    softmax_kernel<<<BB, 256, 0, stream>>>(mask, scores);

    // 3) partial weighted sums over T-chunks
    wsum_kernel<<<dim3(SCH, BB), 256, 0, stream>>>(x_temp, scores, partial);

    // 4) fold partials into (B,D) output
    reduce_kernel<<<(BB * DD) / 256, 256, 0, stream>>>(partial, out);
}